// ReinforceAgentQuantum_29875792511859
// MI455X (gfx1250) — compile-verified
//
#include <hip/hip_runtime.h>
#include <cstdint>

// ---------------------------------------------------------------------------
// 16-qubit batched quantum circuit on MI455X (gfx1250, wave32).
// State (256 x 65536 complex64 = 128MB) lives in d_ws as separate re/im planes.
// Per layer: fused diagonal-phase + two WMMA passes, each applying two
// 4-qubit tensor-product unitaries (16x16 complex GEMMs) via
// V_WMMA_F32_16X16X4_F32, staged through LDS.
// ---------------------------------------------------------------------------

#define NQ     16
#define QDIM   65536
#define NBATCH 256
#define NLAYER 4
#define NFEAT  136        // 16 h + 120 J
#define PI_F   3.14159265358979323846f

typedef __attribute__((ext_vector_type(2))) float v2f;
typedef __attribute__((ext_vector_type(8))) float v8f;

__device__ __forceinline__ int pair_idx(int i, int j) {
  // triu_indices(16,1) linear index for (i<j)
  return i * (2 * NQ - i - 1) / 2 + (j - i - 1);
}

// ---------------------------------------------------------------------------
// Phase tables: angle(b, k) = Ahi[b][khi] + Alo[b][klo] + sum_q W[b][khi][q]*zlo_q
// (input_scaling is linear -> applied at use; tables are layer independent)
// ---------------------------------------------------------------------------
__global__ __launch_bounds__(256)
void qc_precompute(const float* __restrict__ x,
                   float* __restrict__ Ahi, float* __restrict__ Alo,
                   float* __restrict__ W) {
  const int b = blockIdx.x, t = threadIdx.x;
  __shared__ float h[16];
  __shared__ float J[120];
  if (t < NFEAT) {
    float v = x[(size_t)b * NFEAT + t];
    if (t < 16) h[t] = v; else J[t - 16] = v;
  }
  __syncthreads();
  float zh[8];
  #pragma unroll
  for (int q = 0; q < 8; ++q) zh[q] = ((t >> (7 - q)) & 1) ? -1.0f : 1.0f;

  // hi-half: qubits 0..7 live in bits 15..8 of k -> bit (7-q) of khi
  float ah = 0.0f;
  #pragma unroll
  for (int q = 0; q < 8; ++q) ah += h[q] * zh[q];
  #pragma unroll
  for (int i = 0; i < 8; ++i)
    #pragma unroll
    for (int j = i + 1; j < 8; ++j)
      ah += J[pair_idx(i, j)] * zh[i] * zh[j];
  Ahi[b * 256 + t] = ah;

  // lo-half: qubits 8..15 live in bits 7..0 of k -> bit (7-q) of klo
  float al = 0.0f;
  #pragma unroll
  for (int q = 0; q < 8; ++q) al += h[8 + q] * zh[q];
  #pragma unroll
  for (int i = 8; i < 16; ++i)
    #pragma unroll
    for (int j = i + 1; j < 16; ++j)
      al += J[pair_idx(i, j)] * zh[i - 8] * zh[j - 8];
  Alo[b * 256 + t] = al;

  // cross couplings folded per-khi
  #pragma unroll
  for (int q = 0; q < 8; ++q) {
    float acc = 0.0f;
    #pragma unroll
    for (int i = 0; i < 8; ++i) acc += J[pair_idx(i, 8 + q)] * zh[i];
    W[((size_t)b * 256 + t) * 8 + q] = acc;
  }
}

// ---------------------------------------------------------------------------
// Per-lane B operand for the 4-qubit group unitary U (16x16, entries pure
// real or pure imaginary).  Nibble bit p corresponds to qubit (QBASE - p).
// B layout (4x16 f32, 2 VGPRs, mirrored from A/C ISA layout):
//   VGPR0: K = 0 (lanes<16) / 2 (lanes>=16);  VGPR1: K = 1 / 3
// so for k-chunk t the lane supplies j = 4t + 2*half and j+1.
// ---------------------------------------------------------------------------
template <int QBASE>
__device__ __forceinline__ void make_b(const float* __restrict__ xb,
                                       int n, int half, float cw, float sw,
                                       v2f br[4], v2f bi[4], v2f bni[4]) {
  float c[4], s[4];
  #pragma unroll
  for (int p = 0; p < 4; ++p) {
    const bool act = xb[QBASE - p] > 0.0f;   // ann = pi iff h > 0
    c[p] = act ? cw : 1.0f;
    s[p] = act ? sw : 0.0f;
  }
  #pragma unroll
  for (int t = 0; t < 4; ++t) {
    #pragma unroll
    for (int u = 0; u < 2; ++u) {
      const int j = 4 * t + 2 * half + u;
      const int d = n ^ j;
      float mag = 1.0f;
      #pragma unroll
      for (int p = 0; p < 4; ++p) mag *= ((d >> p) & 1) ? s[p] : c[p];
      const int pc = __popc(d);                 // phase (-i)^pc
      const float ur = (pc == 0) ? mag : ((pc == 2) ? -mag : 0.0f);
      const float ui = (pc == 1) ? -mag : ((pc == 3) ? mag : 0.0f);
      if (u == 0) { br[t].x = ur; bi[t].x = ui; bni[t].x = -ui; }
      else        { br[t].y = ur; bi[t].y = ui; bni[t].y = -ui; }
    }
  }
}

// ---------------------------------------------------------------------------
// One 16x16 complex tile: S'(m, n) = sum_j S(m, j) * U(n, j)   (D = S U^T)
// A columns at (colbase + j*colstride) in LDS; writeback to same column set.
// Re = Sr Ur^T - Si Ui^T ;  Im = Sr Ui^T + Si Ur^T  -> 16 wmma_f32_16x16x4.
// ---------------------------------------------------------------------------
__device__ __forceinline__ void apply_tile(float (* __restrict__ sre)[257],
                                           float (* __restrict__ sim)[257],
                                           int colbase, int colstride,
                                           int half, int n,
                                           const v2f br[4], const v2f bi[4],
                                           const v2f bni[4]) {
  v8f dre = {};
  v8f dim_ = {};
  #pragma unroll
  for (int t = 0; t < 4; ++t) {
    const int jx = 4 * t + 2 * half;           // A layout: K = 2*half, 2*half+1
    const int cx = colbase + jx * colstride;
    const int cy = cx + colstride;
    v2f ar, ai;
    ar.x = sre[n][cx]; ar.y = sre[n][cy];      // lane m == n == lane&15
    ai.x = sim[n][cx]; ai.y = sim[n][cy];
    dre  = __builtin_amdgcn_wmma_f32_16x16x4_f32(false, ar, false, br[t],  (short)0, dre,  false, false);
    dre  = __builtin_amdgcn_wmma_f32_16x16x4_f32(false, ai, false, bni[t], (short)0, dre,  false, false);
    dim_ = __builtin_amdgcn_wmma_f32_16x16x4_f32(false, ar, false, bi[t],  (short)0, dim_, false, false);
    dim_ = __builtin_amdgcn_wmma_f32_16x16x4_f32(false, ai, false, br[t],  (short)0, dim_, false, false);
  }
  const int oc = colbase + n * colstride;
  #pragma unroll
  for (int v = 0; v < 8; ++v) {                // D: VGPR v -> row v + 8*half
    const int row = v + 8 * half;
    sre[row][oc] = dre[v];
    sim[row][oc] = dim_[v];
  }
}

// ---------------------------------------------------------------------------
// Layer pass. COLS_LOW: tile columns = k bits 7..0 (pass A, contiguous,
// carries the fused diagonal phase and optional init); else columns =
// k bits 15..8 (pass B).  Applies the two 4-qubit groups living in the
// column byte.  Workgroup = 16 rows x 256 cols complex in LDS, 8 waves,
// each wave: 2 tiles per group, 16 WMMAs per tile.
// ---------------------------------------------------------------------------
template <bool COLS_LOW, bool DO_PHASE, bool INIT, int QBASE_LO, int QBASE_MID>
__global__ __launch_bounds__(256)
void qc_pass(float* __restrict__ gre, float* __restrict__ gim,
             const float* __restrict__ x,
             const float* __restrict__ Ahi, const float* __restrict__ Alo,
             const float* __restrict__ W,
             const float* __restrict__ isc, const float* __restrict__ wts,
             int layer) {
  __shared__ float sre[16][257];   // +1 pad -> conflict-free strided reads
  __shared__ float sim[16][257];
  const int b   = blockIdx.y;
  const int rb  = blockIdx.x;
  const int tid = threadIdx.x;
  const size_t base = (size_t)b * QDIM;

  // ---- load (+ fused phase / init) ----
  if (COLS_LOW) {
    const size_t kb = base + (size_t)rb * 4096;     // fully contiguous block
    const float s = isc[layer];
    for (int e = tid; e < 4096; e += 256) {
      const int row = e >> 8, col = e & 255;
      float re, im;
      if (INIT) { re = 0.00390625f; im = 0.0f; }    // 1/sqrt(DIM) = 1/256
      else      { re = gre[kb + e]; im = gim[kb + e]; }
      if (DO_PHASE) {
        const int khi = rb * 16 + row;
        const float* wr = W + ((size_t)b * 256 + khi) * 8;
        float ang = Ahi[b * 256 + khi] + Alo[b * 256 + col];
        #pragma unroll
        for (int q = 0; q < 8; ++q) {
          const float wq = wr[q];
          ang += ((col >> (7 - q)) & 1) ? -wq : wq;
        }
        const float ha = -0.5f * s * ang;
        float sp, cp;
        __sincosf(ha, &sp, &cp);
        const float nr = re * cp - im * sp;
        const float ni = re * sp + im * cp;
        re = nr; im = ni;
      }
      sre[row][col] = re;
      sim[row][col] = im;
    }
  } else {
    for (int e = tid; e < 4096; e += 256) {         // 64B-contiguous runs
      const int col = e >> 4, row = e & 15;
      const size_t k = base + (size_t)col * 256 + (size_t)rb * 16 + row;
      sre[row][col] = gre[k];
      sim[row][col] = gim[k];
    }
  }
  __syncthreads();

  // ---- group unitaries (RX is identity or RX(w*pi) per qubit) ----
  const float th = 0.5f * wts[layer] * PI_F;
  const float cw = __cosf(th);
  const float sw = __sinf(th);
  const int lane = tid & 31;
  const int wv   = tid >> 5;
  const int half = lane >> 4;
  const int n    = lane & 15;
  const float* xb = x + (size_t)b * NFEAT;

  {   // low nibble of column byte: columns cm*16 + j, stride 1
    v2f br[4], bi[4], bni[4];
    make_b<QBASE_LO>(xb, n, half, cw, sw, br, bi, bni);
    #pragma unroll
    for (int ti = 0; ti < 2; ++ti)
      apply_tile(sre, sim, (wv * 2 + ti) * 16, 1, half, n, br, bi, bni);
  }
  __syncthreads();
  {   // high nibble of column byte: columns j*16 + cl, stride 16
    v2f br[4], bi[4], bni[4];
    make_b<QBASE_MID>(xb, n, half, cw, sw, br, bi, bni);
    #pragma unroll
    for (int ti = 0; ti < 2; ++ti)
      apply_tile(sre, sim, wv * 2 + ti, 16, half, n, br, bi, bni);
  }
  __syncthreads();

  // ---- store ----
  if (COLS_LOW) {
    const size_t kb = base + (size_t)rb * 4096;
    for (int e = tid; e < 4096; e += 256) {
      gre[kb + e] = sre[e >> 8][e & 255];
      gim[kb + e] = sim[e >> 8][e & 255];
    }
  } else {
    for (int e = tid; e < 4096; e += 256) {
      const int col = e >> 4, row = e & 15;
      const size_t k = base + (size_t)col * 256 + (size_t)rb * 16 + row;
      gre[k] = sre[row][col];
      gim[k] = sim[row][col];
    }
  }
}

// ---------------------------------------------------------------------------
// probs -> <Z_a> (a = 0..15) * output_scaling.  Thread t owns klo = t, so
// actions 8..15 have a fixed per-thread sign; actions 0..7 sign varies with
// the khi loop only.
// ---------------------------------------------------------------------------
__global__ __launch_bounds__(256)
void qc_final(const float* __restrict__ gre, const float* __restrict__ gim,
              const float* __restrict__ osc, float* __restrict__ out) {
  const int b = blockIdx.x, t = threadIdx.x;
  const float* re = gre + (size_t)b * QDIM;
  const float* im = gim + (size_t)b * QDIM;
  float T = 0.0f;
  float U[8] = {};
  for (int ktop = 0; ktop < 256; ++ktop) {
    const int k = ktop * 256 + t;
    const float r = re[k], i = im[k];
    const float p = __fmaf_rn(r, r, i * i);
    T += p;
    #pragma unroll
    for (int a = 0; a < 8; ++a)
      U[a] += ((ktop >> (7 - a)) & 1) ? -p : p;
  }
  __shared__ float acc[256];
  __shared__ float Sa[16];
  for (int a = 0; a < 16; ++a) {
    float v;
    if (a < 8) v = U[a];
    else       v = (((t >> (15 - a)) & 1) ? -T : T);
    acc[t] = v;
    __syncthreads();
    for (int off = 128; off > 0; off >>= 1) {
      if (t < off) acc[t] += acc[t + off];
      __syncthreads();
    }
    if (t == 0) Sa[a] = acc[0];
    __syncthreads();
  }
  if (t < 16) out[b * 16 + t] = Sa[t] * osc[t];
}

// ---------------------------------------------------------------------------
extern "C" void kernel_launch(void* const* d_in, const int* in_sizes, int n_in,
                              void* d_out, int out_size, void* d_ws, size_t ws_size,
                              hipStream_t stream) {
  const float* x   = (const float*)d_in[0];   // (256, 136)
  const float* isc = (const float*)d_in[1];   // (4,)
  const float* wts = (const float*)d_in[2];   // (4,)
  const float* osc = (const float*)d_in[3];   // (16,)
  float* out = (float*)d_out;                 // (256, 16)

  char* ws = (char*)d_ws;
  const size_t state_bytes = (size_t)NBATCH * QDIM * sizeof(float);   // 64MB each
  float* sre = (float*)(ws);
  float* sim = (float*)(ws + state_bytes);
  float* Ahi = (float*)(ws + 2 * state_bytes);
  float* Alo = (float*)(ws + 2 * state_bytes + (size_t)NBATCH * 256 * 4);
  float* W   = (float*)(ws + 2 * state_bytes + (size_t)NBATCH * 256 * 8);
  (void)in_sizes; (void)n_in; (void)out_size; (void)ws_size;

  qc_precompute<<<NBATCH, 256, 0, stream>>>(x, Ahi, Alo, W);

  const dim3 grid(16, NBATCH);
  for (int L = 0; L < NLAYER; ++L) {
    // pass A: cols = k bits 7..0 -> qubits 15..12 (nibble lo, qbase 15)
    //                              and 11..8 (nibble hi, qbase 11); fused phase
    if (L == 0)
      qc_pass<true,  true,  true,  15, 11><<<grid, 256, 0, stream>>>(sre, sim, x, Ahi, Alo, W, isc, wts, L);
    else
      qc_pass<true,  true,  false, 15, 11><<<grid, 256, 0, stream>>>(sre, sim, x, Ahi, Alo, W, isc, wts, L);
    // pass B: cols = k bits 15..8 -> qubits 7..4 (qbase 7) and 3..0 (qbase 3)
    qc_pass<false, false, false, 7, 3><<<grid, 256, 0, stream>>>(sre, sim, x, Ahi, Alo, W, isc, wts, L);
  }

  qc_final<<<NBATCH, 256, 0, stream>>>(sre, sim, osc, out);
}